// CommAwareGAT_39496519254635
// MI455X (gfx1250) — compile-verified
//
#include <hip/hip_runtime.h>
#include <hip/hip_bf16.h>

// CDNA5 / gfx1250, wave32.
typedef __attribute__((ext_vector_type(2))) float v2f;
typedef __attribute__((ext_vector_type(8))) float v8f;

#define IN_C  128
#define OUT_C 64

// ---------------------------------------------------------------------------
// Kernel 1: fused h = x@W_conv ; out = x@W_res + bias  (fp32 WMMA 16x16x4)
// One wave computes one 16-row tile of both outputs.
// Block = 256 threads = 8 waves -> 128 rows per block.
//
// LDS weight layout is fragment-interleaved: element W[k][col] lives at
//   sW[(k>>2)*256 + col*4 + (k&2) + (k&1)]
// so a B fragment for (kb, khalf, col) = {W[4kb+2kh][col], W[4kb+2kh+1][col]}
// is one aligned 8-byte LDS load, and the 32 lanes of a wave touch all 64
// banks exactly once (conflict-free).
// ---------------------------------------------------------------------------
__global__ __launch_bounds__(256)
void gat_gemm_kernel(const float* __restrict__ x,
                     const float* __restrict__ Wc,
                     const float* __restrict__ Wr,
                     const float* __restrict__ bias,
                     float* __restrict__ h,
                     float* __restrict__ out,
                     int N)
{
    __shared__ float sWc[IN_C * OUT_C];   // 32 KB
    __shared__ float sWr[IN_C * OUT_C];   // 32 KB

    // Cooperative stage of both weight matrices into LDS, interleaved.
    for (int i = threadIdx.x; i < IN_C * OUT_C; i += 256) {
        const int k   = i >> 6;           // 0..127
        const int col = i & 63;           // 0..63
        const int idx = ((k >> 2) << 8) + (col << 2) + (k & 2) + (k & 1);
        sWc[idx] = Wc[i];
        sWr[idx] = Wr[i];
    }
    __syncthreads();

    const int lane = threadIdx.x & 31;
    const int wave = threadIdx.x >> 5;
    const int m    = lane & 15;   // row (A) / col (B,C,D) within 16
    const int kh   = lane >> 4;   // which K-pair half of the wave

    const int rowbase = (blockIdx.x * 8 + wave) * 16;
    if (rowbase >= N) return;                 // wave-uniform: EXEC stays full

    // Per-lane bias values for the 4 N-tiles.
    float bv[4];
#pragma unroll
    for (int nt = 0; nt < 4; ++nt) bv[nt] = bias[nt * 16 + m];

    v8f acc_h[4] = {};
    v8f acc_r[4] = {};

    const float* xrow     = x + (size_t)(rowbase + m) * IN_C;
    const int    fragbase = (m << 2) + (kh << 1);

    // K = 128 in 32 steps of 4.
    for (int kb = 0; kb < 32; ++kb) {
        // A fragment (16x4 fp32): lane holds A[m][k0], A[m][k0+1].
        v2f a = *(const v2f*)(xrow + kb * 4 + kh * 2);

        const float* pc = sWc + kb * 256 + fragbase;
        const float* pr = sWr + kb * 256 + fragbase;

#pragma unroll
        for (int nt = 0; nt < 4; ++nt) {
            v2f bc = *(const v2f*)(pc + nt * 64);
            v2f br = *(const v2f*)(pr + nt * 64);

            acc_h[nt] = __builtin_amdgcn_wmma_f32_16x16x4_f32(
                false, a, false, bc, (short)0, acc_h[nt], false, false);
            acc_r[nt] = __builtin_amdgcn_wmma_f32_16x16x4_f32(
                false, a, false, br, (short)0, acc_r[nt], false, false);
        }
    }

    // C/D layout: VGPR v -> row (v + 8*kh), col = lane&15 within the N-tile.
#pragma unroll
    for (int nt = 0; nt < 4; ++nt) {
#pragma unroll
        for (int v = 0; v < 8; ++v) {
            const int r = rowbase + kh * 8 + v;
            const int c = nt * 16 + m;
            h  [(size_t)r * OUT_C + c] = acc_h[nt][v];
            out[(size_t)r * OUT_C + c] = acc_r[nt][v] + bv[nt];
        }
    }
}

// ---------------------------------------------------------------------------
// Kernel 2: per-edge attention numerator + scatter-sum of denominators.
// One wave per edge: 32 lanes x float2 = 64 columns.
// Edge index is forced scalar so src/dst loads can use the SMEM path.
// ---------------------------------------------------------------------------
__global__ __launch_bounds__(256)
void gat_score_kernel(const float* __restrict__ h,
                      const int*   __restrict__ ei,   // [2][E]
                      const float* __restrict__ Watt, // [128]
                      const float* __restrict__ batt, // [1]
                      float* __restrict__ num,        // [E]
                      float* __restrict__ denom,      // [N]
                      int E)
{
    const int lane = threadIdx.x & 31;
    const int e = __builtin_amdgcn_readfirstlane(blockIdx.x * 8 + (threadIdx.x >> 5));
    if (e >= E) return;                        // wave-uniform

    const int src = ei[e];
    const int dst = ei[E + e];
    const int c   = lane * 2;

    const float2 hi = *(const float2*)(h + (size_t)dst * OUT_C + c);
    const float2 hj = *(const float2*)(h + (size_t)src * OUT_C + c);
    const float2 wi = *(const float2*)(Watt + c);
    const float2 wj = *(const float2*)(Watt + OUT_C + c);

    float p = hi.x * wi.x + hi.y * wi.y + hj.x * wj.x + hj.y * wj.y;

    // wave32 butterfly reduction
#pragma unroll
    for (int off = 16; off > 0; off >>= 1)
        p += __shfl_xor(p, off, 32);

    if (lane == 0) {
        float s = p + batt[0];
        s = (s > 0.0f) ? s : 0.2f * s;         // leaky_relu(0.2)
        const float nu = __expf(s);
        num[e] = nu;
        atomicAdd(denom + dst, nu);
    }
}

// ---------------------------------------------------------------------------
// Kernel 3: alpha-weighted scatter-aggregate of source features.
// ---------------------------------------------------------------------------
__global__ __launch_bounds__(256)
void gat_aggr_kernel(const float* __restrict__ h,
                     const int*   __restrict__ ei,
                     const float* __restrict__ num,
                     const float* __restrict__ denom,
                     float* __restrict__ out,
                     int E)
{
    const int lane = threadIdx.x & 31;
    const int e = __builtin_amdgcn_readfirstlane(blockIdx.x * 8 + (threadIdx.x >> 5));
    if (e >= E) return;

    const int src = ei[e];
    const int dst = ei[E + e];

    const float alpha = num[e] / (denom[dst] + 1e-16f);

    const int c = lane * 2;
    const float2 hj = *(const float2*)(h + (size_t)src * OUT_C + c);

    atomicAdd(out + (size_t)dst * OUT_C + c,     alpha * hj.x);
    atomicAdd(out + (size_t)dst * OUT_C + c + 1, alpha * hj.y);
}

// ---------------------------------------------------------------------------
// Launch
// ---------------------------------------------------------------------------
extern "C" void kernel_launch(void* const* d_in, const int* in_sizes, int n_in,
                              void* d_out, int out_size, void* d_ws, size_t ws_size,
                              hipStream_t stream)
{
    const float* x    = (const float*)d_in[0];
    const int*   ei   = (const int*)  d_in[1];
    const float* Wc   = (const float*)d_in[2];
    const float* Watt = (const float*)d_in[3];
    const float* batt = (const float*)d_in[4];
    const float* Wr   = (const float*)d_in[5];
    const float* bias = (const float*)d_in[6];

    const int N = in_sizes[0] / IN_C;
    const int E = in_sizes[1] / 2;

    float* out   = (float*)d_out;
    float* h     = (float*)d_ws;            // N * 64 floats
    float* num   = h + (size_t)N * OUT_C;   // E floats
    float* denom = num + (size_t)E;         // N floats

    // denominators must start at zero every call (ws is not re-poisoned).
    hipMemsetAsync(denom, 0, (size_t)N * sizeof(float), stream);

    const dim3 blk(256);
    const int g1 = (N + 127) / 128;         // 8 waves * 16 rows per block
    gat_gemm_kernel<<<g1, blk, 0, stream>>>(x, Wc, Wr, bias, h, out, N);

    const int g2 = (E + 7) / 8;             // 8 waves (edges) per block
    gat_score_kernel<<<g2, blk, 0, stream>>>(h, ei, Watt, batt, num, denom, E);
    gat_aggr_kernel <<<g2, blk, 0, stream>>>(h, ei, num, denom, out, E);
}